// SpikeFP64Sin_43860206027304
// MI455X (gfx1250) — compile-verified
//
#include <hip/hip_runtime.h>
#include <stdint.h>

// ---------------------------------------------------------------------------
// SpikeFP64Sin on bit-expanded f64 tensors (gfx1250).
// Input : N*64 float32 "bits" (MSB-first IEEE754 f64 layout)
// Output: N*64 float32 "bits" of sin(x) per the reference circuit.
//
// Memory-bound (512 MB @ 23.3 TB/s ~ 22 us). Strategy:
//   global --(global_load_async_to_lds_b128, coalesced)--> padded LDS rows
//   per-lane ds_load_b128 -> pack 64 bits -> f64 sin circuit -> unpack
//   padded LDS rows --(global_store_async_from_lds_b128, coalesced)--> global
// Full blocks take an unguarded straight-line async clause (no exec-mask
// churn); only a tail block (never hit at 1024x1024) takes guarded loops.
// ---------------------------------------------------------------------------

#define EPB   128           // elements per block (1 per thread)
#define ROWB  272           // padded LDS row stride (256B data + 16B pad)
                            // lane l row starts at bank (4*l)%64 -> minimal conflicts

__device__ __forceinline__ unsigned long long spike_sin_bits(unsigned long long xu)
{
#pragma clang fp contract(off)
  const double PI          = 3.141592653589793;     // == math.pi
  const double TWO_OVER_PI = 2.0 / PI;              // compile-time double == python
  const double PI_OVER_2   = PI / 2.0;

  const double x = __longlong_as_double((long long)xu);

  // q = SpikeFP64Round(x * 2/pi): floor(|xq|+0.5), reattach xq's sign bit
  double xq = x * TWO_OVER_PI;
  unsigned long long xqu = (unsigned long long)__double_as_longlong(xq);
  bool qneg = (xqu >> 63) & 1ull;
  double qa = floor(fabs(xq) + 0.5);
  double q  = qneg ? -qa : qa;                      // preserves -0 like jnp.where
  double qp = q * PI_OVER_2;
  double r  = x + (-qp);                            // circuit: flip sign bit, add

  // SpikeFP64ExtractMod4 on q's raw bits
  unsigned long long qu = (unsigned long long)__double_as_longlong(q);
  bool e10 = (qu >> 62) & 1ull;                     // exponent MSB
  bool e0  = (qu >> 52) & 1ull;                     // exponent LSB
  bool m51 = (qu >> 51) & 1ull;                     // mantissa MSB
  bool m50 = (qu >> 50) & 1ull;
  bool b0h = e0 ? m50 : m51;
  bool b1h = e0 ? m51 : true;
  bool b0  = e10 ? b0h : e0;
  bool b1  = e10 ? b1h : false;
  if (qneg) b1 = b1 ^ b0;                           // sign correction for mod 4

  // Taylor polys, same op/sum order as reference (no FMA contraction)
  double r2 = r * r;
  double r3 = r2 * r;
  double r5 = r3 * r2;
  double r7 = r5 * r2;
  double sin_r = ((r + r3 * (-1.0 / 6.0)) + r5 * (1.0 / 120.0)) + r7 * (-1.0 / 5040.0);
  double r4 = r2 * r2;
  double r6 = r4 * r2;
  double cos_r = ((1.0 + r2 * (-0.5)) + r4 * (1.0 / 24.0)) + r6 * (-1.0 / 720.0);

  double base = b0 ? cos_r : sin_r;
  unsigned long long bu = (unsigned long long)__double_as_longlong(base);
  unsigned long long osign = ((bu >> 63) ^ (unsigned long long)b1) & 1ull;
  return (bu & 0x7FFFFFFFFFFFFFFFull) | (osign << 63);
}

// per-k tile geometry: byte g = k*2048 + tid*16; element e = g>>8; w = g&255
__device__ __forceinline__ void async_tile_in(const unsigned char* gin,
                                              uint32_t lbase, unsigned tid)
{
#pragma unroll
  for (int k = 0; k < 16; ++k) {
    uint32_t g = (uint32_t)k * (EPB * 16u) + tid * 16u;
    uint32_t laddr = lbase + (g >> 8) * ROWB + (g & 255u);
    unsigned long long gaddr = (unsigned long long)(uintptr_t)(gin + g);
    asm volatile("global_load_async_to_lds_b128 %0, %1, off"
                 :: "v"(laddr), "v"(gaddr) : "memory");
  }
}

__device__ __forceinline__ void async_tile_out(unsigned char* gout,
                                               uint32_t lbase, unsigned tid)
{
#pragma unroll
  for (int k = 0; k < 16; ++k) {
    uint32_t g = (uint32_t)k * (EPB * 16u) + tid * 16u;
    uint32_t laddr = lbase + (g >> 8) * ROWB + (g & 255u);
    unsigned long long gaddr = (unsigned long long)(uintptr_t)(gout + g);
    asm volatile("global_store_async_from_lds_b128 %1, %0, off"
                 :: "v"(laddr), "v"(gaddr) : "memory");
  }
}

__global__ __launch_bounds__(EPB) void spike_sin_kernel(
    const float* __restrict__ in_bits, float* __restrict__ out_bits,
    unsigned long long nelem)
{
  __shared__ __align__(16) unsigned char smem[EPB * ROWB];   // 34816 B

  const unsigned tid = threadIdx.x;
  const unsigned long long blockBase = (unsigned long long)blockIdx.x * EPB;
  const unsigned long long elem = blockBase + tid;
  const bool fullBlock = (blockBase + EPB) <= nelem;          // uniform per block

  const unsigned char* gin  = (const unsigned char*)in_bits + blockBase * 256ull;
  unsigned char*       gout = (unsigned char*)out_bits      + blockBase * 256ull;
  // generic->LDS: low 32 bits of a generic LDS pointer are the LDS byte offset
  const uint32_t lbase = (uint32_t)(uintptr_t)(void*)smem;

  // ---- Phase 1: async coalesced global -> padded LDS (32 KB tile) ----------
  if (fullBlock) {
    async_tile_in(gin, lbase, tid);                 // straight-line, no exec churn
  } else {
#pragma unroll
    for (int k = 0; k < 16; ++k) {                  // tail block only
      uint32_t g = (uint32_t)k * (EPB * 16u) + tid * 16u;
      uint32_t e = g >> 8;
      if (blockBase + e < nelem) {
        uint32_t laddr = lbase + e * ROWB + (g & 255u);
        unsigned long long gaddr = (unsigned long long)(uintptr_t)(gin + g);
        asm volatile("global_load_async_to_lds_b128 %0, %1, off"
                     :: "v"(laddr), "v"(gaddr) : "memory");
      }
    }
  }
  asm volatile("s_wait_asynccnt 0x0" ::: "memory");
  __syncthreads();

  // ---- Phase 2-4: pack bits, sin circuit, unpack bits ----------------------
  if (elem < nelem) {
    const uint32_t row = tid * (uint32_t)ROWB;

    uint32_t hi = 0u, lo = 0u;
#pragma unroll
    for (int j = 0; j < 8; ++j) {       // bits 0..31 -> u64[63:32]
      float4 v = *(const float4*)(smem + row + j * 16);
      hi = (hi << 1) | ((__float_as_uint(v.x) >> 29) & 1u);  // 1.0f->1, 0.0f->0
      hi = (hi << 1) | ((__float_as_uint(v.y) >> 29) & 1u);
      hi = (hi << 1) | ((__float_as_uint(v.z) >> 29) & 1u);
      hi = (hi << 1) | ((__float_as_uint(v.w) >> 29) & 1u);
    }
#pragma unroll
    for (int j = 8; j < 16; ++j) {      // bits 32..63 -> u64[31:0]
      float4 v = *(const float4*)(smem + row + j * 16);
      lo = (lo << 1) | ((__float_as_uint(v.x) >> 29) & 1u);
      lo = (lo << 1) | ((__float_as_uint(v.y) >> 29) & 1u);
      lo = (lo << 1) | ((__float_as_uint(v.z) >> 29) & 1u);
      lo = (lo << 1) | ((__float_as_uint(v.w) >> 29) & 1u);
    }
    unsigned long long xu = ((unsigned long long)hi << 32) | (unsigned long long)lo;

    unsigned long long ou = spike_sin_bits(xu);

    const uint32_t ohi = (uint32_t)(ou >> 32);
    const uint32_t olo = (uint32_t)ou;
#pragma unroll
    for (int j = 0; j < 16; ++j) {      // MSB-first unpack into same LDS row
      uint32_t wsrc = (j < 8) ? ohi : olo;
      int base = 31 - 4 * (j & 7);
      float4 o;
      o.x = (float)((wsrc >> (base - 0)) & 1u);
      o.y = (float)((wsrc >> (base - 1)) & 1u);
      o.z = (float)((wsrc >> (base - 2)) & 1u);
      o.w = (float)((wsrc >> (base - 3)) & 1u);
      *(float4*)(smem + row + j * 16) = o;
    }
  }
  __syncthreads();   // dscnt wait + barrier: all rows written before async store

  // ---- Phase 5: async coalesced padded LDS -> global -----------------------
  if (fullBlock) {
    async_tile_out(gout, lbase, tid);
  } else {
#pragma unroll
    for (int k = 0; k < 16; ++k) {                  // tail block only
      uint32_t g = (uint32_t)k * (EPB * 16u) + tid * 16u;
      uint32_t e = g >> 8;
      if (blockBase + e < nelem) {
        uint32_t laddr = lbase + e * ROWB + (g & 255u);
        unsigned long long gaddr = (unsigned long long)(uintptr_t)(gout + g);
        asm volatile("global_store_async_from_lds_b128 %1, %0, off"
                     :: "v"(laddr), "v"(gaddr) : "memory");
      }
    }
  }
  asm volatile("s_wait_asynccnt 0x0" ::: "memory");
}

extern "C" void kernel_launch(void* const* d_in, const int* in_sizes, int n_in,
                              void* d_out, int out_size, void* d_ws, size_t ws_size,
                              hipStream_t stream) {
  (void)n_in; (void)d_ws; (void)ws_size; (void)out_size;
  const float* x = (const float*)d_in[0];
  float* y = (float*)d_out;
  unsigned long long nbits = (unsigned long long)in_sizes[0];
  unsigned long long nelem = nbits / 64ull;                 // 1,048,576 for 1024x1024
  unsigned long long blocks = (nelem + EPB - 1) / EPB;      // 8192 blocks
  spike_sin_kernel<<<(dim3)(unsigned)blocks, EPB, 0, stream>>>(x, y, nelem);
}